// ProtectedMemModel_77068893160208
// MI455X (gfx1250) — compile-verified
//
#include <hip/hip_runtime.h>
#include <hip/hip_bf16.h>

// CDNA5 / gfx1250, wave32. Full-f32 WMMA (16x16x4) to preserve the reference's
// f32 ranking semantics in the gate and the f32 readout.

typedef float v2f __attribute__((ext_vector_type(2)));
typedef float v4f __attribute__((ext_vector_type(4)));
typedef float v8f __attribute__((ext_vector_type(8)));

#define HID 64
#define SLOTS 8
#define SEQL 24
#define TSTEPS 23
#define VOC 64
#define BATCH 16384

__device__ __forceinline__ v8f wmma_f32(v2f a, v2f b, v8f c) {
  // 8-arg form: (neg_a, A, neg_b, B, c_mod, C, reuse_a, reuse_b)
  return __builtin_amdgcn_wmma_f32_16x16x4_f32(false, a, false, b, (short)0, c,
                                               false, false);
}

// ---------------------------------------------------------------------------
// Kernel 1: the 23-step protected-memory scan. One wave = 2 batches stacked
// into one 16-row WMMA A-matrix. Block = 256 threads = 8 waves = 16 batches.
// Token/embedding fetches are software-pipelined 2 deep so their latency hides
// behind the 32 WMMAs of the gate instead of stalling at the loop top.
// Writes mem_summary (B x 64 f32) to workspace.
// ---------------------------------------------------------------------------
__global__ __launch_bounds__(256) void scan_kernel(
    const int* __restrict__ seqs, const float* __restrict__ embed,
    const float* __restrict__ gw1, const float* __restrict__ gb1,
    const float* __restrict__ gw2, const float* __restrict__ gb2,
    float* __restrict__ summary) {
  // [wave][row 0-7 = batch0 slots, 8-15 = batch1 slots][dim]
  __shared__ __align__(16) float mem[8][16][HID];  // 32 KB

  const int tid = threadIdx.x;
  const int wave = tid >> 5;
  const int lane = tid & 31;
  const int laneq = lane & 15;  // row (A) / column (B,C) index
  const int half = lane >> 4;   // 0: batch0, 1: batch1
  const int koff = half * 2;    // K sub-offset for f32 WMMA A/B layout
  const int b = blockIdx.x * 16 + wave * 2 + half;
  float* M = &mem[wave][0][0];

  // Preload gate_w1 (64x32) into B-layout registers: 2 N-tiles x 16 K-chunks.
  // B 4x16 layout: lanes 0-15 hold K = {4c, 4c+1}, lanes 16-31 K = {4c+2,
  // 4c+3}, column N = laneq (+16 per tile).
  v2f Bw1[2][16];
#pragma unroll
  for (int nt = 0; nt < 2; ++nt) {
#pragma unroll
    for (int c = 0; c < 16; ++c) {
      const int k = 4 * c + koff;
      const int n = nt * 16 + laneq;
      Bw1[nt][c].x = gw1[k * 32 + n];
      Bw1[nt][c].y = gw1[(k + 1) * 32 + n];
    }
  }
  const float b1v0 = gb1[laneq];
  const float b1v1 = gb1[16 + laneq];
  const float w2v0 = gw2[laneq];
  const float w2v1 = gw2[16 + laneq];
  const float b2v = gb2[0];

  const int rowbase = half * 8;  // this half's batch rows in LDS
  const int dbase = laneq * 4;   // 4 contiguous dims per lane (float4)
  const int* seqrow = &seqs[b * SEQL];

  // 2-deep software pipeline: tokNext resident one step early, embCur gather
  // for step t issued during step t-1.
  int tokCur = seqrow[0];
  int tokNext = seqrow[1];
  v4f embCur = *(const v4f*)&embed[tokCur * HID + dbase];

  for (int t = 0; t < TSTEPS; ++t) {
    // Issue next-step prefetches first; their waits land after the gate WMMAs.
    const int pidx = (t + 2 <= SEQL - 1) ? (t + 2) : (SEQL - 1);
    const int tokNext2 = seqrow[pidx];
    const v4f embNext = *(const v4f*)&embed[tokNext * HID + dbase];

    if (t < SLOTS) {
      // fill phase: slot t := emb
      *(v4f*)&M[(rowbase + t) * HID + dbase] = embCur;
    } else {
      // ---- gate scores on pre-update mem via f32 WMMA ----
      // A chunks: lane reads row = laneq, K = {4c+koff, 4c+koff+1}
      v2f A[16];
#pragma unroll
      for (int c = 0; c < 16; ++c)
        A[c] = *(const v2f*)&M[laneq * HID + 4 * c + koff];

      v8f C0, C1;
#pragma unroll
      for (int j = 0; j < 8; ++j) {
        C0[j] = b1v0;
        C1[j] = b1v1;
      }
#pragma unroll
      for (int c = 0; c < 16; ++c) {
        C0 = wmma_f32(A[c], Bw1[0][c], C0);
        C1 = wmma_f32(A[c], Bw1[1][c], C1);
      }

      // relu + dot with gate_w2, reduce across the 16 lanes of each half.
      // C layout: lane = column n; VGPR j = row j (lanes 0-15) / j+8 (16-31),
      // so after the butterfly each half holds its own batch's 8 slot scores.
      float sc[8];
#pragma unroll
      for (int j = 0; j < 8; ++j) {
        const float h0 = C0[j] > 0.f ? C0[j] : 0.f;
        const float h1 = C1[j] > 0.f ? C1[j] : 0.f;
        sc[j] = h0 * w2v0 + h1 * w2v1;
      }
#pragma unroll
      for (int m = 1; m < 16; m <<= 1) {
#pragma unroll
        for (int j = 0; j < 8; ++j) sc[j] += __shfl_xor(sc[j], m, 32);
      }
#pragma unroll
      for (int j = 0; j < 8; ++j) sc[j] += b2v;

      // ranks (stable descending argsort-of-argsort), branch-free
      int rank[8];
#pragma unroll
      for (int s = 0; s < 8; ++s) {
        int r = 0;
#pragma unroll
        for (int u = 0; u < 8; ++u)
          r += (sc[u] > sc[s]) || ((sc[u] == sc[s]) && (u < s));
        rank[s] = r;
      }
      int evict = 7;
#pragma unroll
      for (int s = 6; s >= 0; --s) evict = (rank[s] >= 2) ? s : evict;

      // shift out evicted slot; new emb in slot 7 (read-all then write-all)
      v4f vals[7];
#pragma unroll
      for (int s = 0; s < 7; ++s) {
        const int src = (s < evict) ? s : s + 1;
        vals[s] = *(const v4f*)&M[(rowbase + src) * HID + dbase];
      }
#pragma unroll
      for (int s = 0; s < 7; ++s)
        *(v4f*)&M[(rowbase + s) * HID + dbase] = vals[s];
      *(v4f*)&M[(rowbase + 7) * HID + dbase] = embCur;
    }

    embCur = embNext;
    tokNext = tokNext2;
  }

  // mem_summary = mean over slots
  v4f acc = {0.f, 0.f, 0.f, 0.f};
#pragma unroll
  for (int s = 0; s < SLOTS; ++s) {
    const v4f v = *(const v4f*)&M[(rowbase + s) * HID + dbase];
    acc += v;
  }
  acc *= 0.125f;
  *(v4f*)&summary[b * HID + dbase] = acc;
}

// ---------------------------------------------------------------------------
// Kernel 2: logits = relu([q_emb|summary] @ rw1 + rb1) @ rw2 + rb2.
// 4 waves cooperate on each 16-batch tile (one N-tile each) for 4x the wave
// parallelism of a monolithic per-tile wave. Block = 256 threads = 2 tiles.
// ---------------------------------------------------------------------------
__global__ __launch_bounds__(256) void readout_kernel(
    const int* __restrict__ qtok, const float* __restrict__ embed,
    const float* __restrict__ summary, const float* __restrict__ rw1,
    const float* __restrict__ rb1, const float* __restrict__ rw2,
    const float* __restrict__ rb2, float* __restrict__ out) {
  __shared__ __align__(16) float hid[2][16 * HID];  // 8 KB C->A transpose tile

  const int tid = threadIdx.x;
  const int wave = tid >> 5;
  const int lane = tid & 31;
  const int laneq = lane & 15;
  const int half = lane >> 4;
  const int koff = half * 2;
  const int tileL = wave >> 2;  // local tile (0..1)
  const int nt = wave & 3;      // this wave's N-tile (16 cols)
  const int m0 = (blockIdx.x * 2 + tileL) * 16;  // first batch of this tile
  const int bm = m0 + laneq;                     // A-row batch for this lane
  const int tok = qtok[bm];
  const int n = nt * 16 + laneq;

  // A = h (16 x 128), assembled on the fly: K<64 -> q_emb, else mem_summary.
  v2f A[32];
#pragma unroll
  for (int c = 0; c < 32; ++c) {
    const int k = 4 * c + koff;
    const float* src =
        (k < HID) ? &embed[tok * HID + k] : &summary[bm * HID + (k - HID)];
    A[c] = *(const v2f*)src;
  }

  // GEMM1 (this wave's 16 columns): (16x128)@(128x16) + b1, relu.
  {
    v8f C;
    const float bias = rb1[n];
#pragma unroll
    for (int j = 0; j < 8; ++j) C[j] = bias;
#pragma unroll
    for (int c = 0; c < 32; ++c) {
      const int k = 4 * c + koff;
      v2f Bv;
      Bv.x = rw1[k * 64 + n];
      Bv.y = rw1[(k + 1) * 64 + n];
      C = wmma_f32(A[c], Bv, C);
    }
    // relu + transpose into row-major LDS (C layout: lane = column n,
    // VGPR j = row j for lanes 0-15 / j+8 for lanes 16-31).
#pragma unroll
    for (int j = 0; j < 8; ++j) {
      const float h = C[j] > 0.f ? C[j] : 0.f;
      hid[tileL][(half * 8 + j) * HID + n] = h;
    }
  }

  __syncthreads();  // hidden tile complete across the 4 cooperating waves

  // GEMM2 (this wave's 16 columns): (16x64)@(64x16) + b2 -> logits.
  v2f A2[16];
#pragma unroll
  for (int c = 0; c < 16; ++c)
    A2[c] = *(const v2f*)&hid[tileL][laneq * HID + 4 * c + koff];

  {
    v8f C;
    const float bias = rb2[n];
#pragma unroll
    for (int j = 0; j < 8; ++j) C[j] = bias;
#pragma unroll
    for (int c = 0; c < 16; ++c) {
      const int k = 4 * c + koff;
      v2f Bv;
      Bv.x = rw2[k * 64 + n];
      Bv.y = rw2[(k + 1) * 64 + n];
      C = wmma_f32(A2[c], Bv, C);
    }
#pragma unroll
    for (int j = 0; j < 8; ++j) out[(m0 + half * 8 + j) * VOC + n] = C[j];
  }
}

extern "C" void kernel_launch(void* const* d_in, const int* in_sizes, int n_in,
                              void* d_out, int out_size, void* d_ws,
                              size_t ws_size, hipStream_t stream) {
  const int* seqs = (const int*)d_in[0];
  const int* qtok = (const int*)d_in[1];
  const float* embed = (const float*)d_in[2];
  const float* gw1 = (const float*)d_in[3];
  const float* gb1 = (const float*)d_in[4];
  const float* gw2 = (const float*)d_in[5];
  const float* gb2 = (const float*)d_in[6];
  const float* rw1 = (const float*)d_in[7];
  const float* rb1 = (const float*)d_in[8];
  const float* rw2 = (const float*)d_in[9];
  const float* rb2 = (const float*)d_in[10];
  float* out = (float*)d_out;
  float* summary = (float*)d_ws;  // B x 64 f32 = 4 MB

  // 1024 blocks x 256 threads: 8 waves/block, 2 batches/wave.
  scan_kernel<<<BATCH / 16, 256, 0, stream>>>(seqs, embed, gw1, gb1, gw2, gb2,
                                              summary);
  // 512 blocks x 256 threads: 8 waves/block, 4 waves cooperate per 16-batch
  // tile -> 4096 waves for latency hiding.
  readout_kernel<<<BATCH / 32, 256, 0, stream>>>(qtok, embed, summary, rw1,
                                                 rb1, rw2, rb2, out);
}